// SharedGeneratorOperator_28570122453138
// MI455X (gfx1250) — compile-verified
//
#include <hip/hip_runtime.h>
#include <hip/hip_bf16.h>

typedef __attribute__((ext_vector_type(2))) float v2f;
typedef __attribute__((ext_vector_type(8))) float v8f;

#define F_DIM 64
#define CODE  32
#define KGEN  16
#define HID   64
#define LD    65            // padded LDS row stride (floats) to avoid bank conflicts
#define NTAY  12            // Taylor/Horner order for expm

// Preload the 16 K-step A-fragments (16x4 tiles) for this wave's row-block.
// Both output tiles owned by a wave share the same mt, so one set suffices.
__device__ __forceinline__ void load_afrag(const float* __restrict__ X, v2f af[16]) {
    const int lane = threadIdx.x & 31;
    const int wave = threadIdx.x >> 5;
    const int ml   = lane & 15;
    const int kh   = lane >> 4;
    const int mt   = (wave >> 1) << 4;   // row-block base for tiles 2w, 2w+1
#pragma unroll
    for (int kk = 0; kk < 16; ++kk) {
        const int kb = kk * 4 + kh * 2;
        af[kk].x = X[(mt + ml) * LD + kb];
        af[kk].y = X[(mt + ml) * LD + kb + 1];
    }
}

// 64x64 fp32 matmul D = X @ Y with X supplied as preloaded register fragments.
// 256 threads = 8 waves; wave w computes output tiles 2w and 2w+1 (same mt).
__device__ __forceinline__ void mm64_pre(float* __restrict__ D, const v2f af[16],
                                         const float* __restrict__ Y) {
    const int lane = threadIdx.x & 31;
    const int wave = threadIdx.x >> 5;
    const int ml   = lane & 15;
    const int kh   = lane >> 4;
    const int mt   = (wave >> 1) << 4;
#pragma unroll
    for (int t = 0; t < 2; ++t) {
        const int nt = ((2 * wave + t) & 3) << 4;   // output tile col base
        v8f acc = {};
#pragma unroll
        for (int kk = 0; kk < 16; ++kk) {
            const int kb = kk * 4 + kh * 2;
            v2f bfr;
            bfr.x = Y[kb * LD + nt + ml];
            bfr.y = Y[(kb + 1) * LD + nt + ml];
            acc = __builtin_amdgcn_wmma_f32_16x16x4_f32(
                false, af[kk], false, bfr, (short)0, acc, false, false);
        }
#pragma unroll
        for (int r = 0; r < 8; ++r) {
            D[(mt + r + kh * 8) * LD + nt + ml] = acc[r];
        }
    }
}

// General 64x64 matmul D = X @ Y, fragments loaded from LDS each call.
__device__ __forceinline__ void mm64(float* __restrict__ D,
                                     const float* __restrict__ X,
                                     const float* __restrict__ Y) {
    v2f af[16];
    load_afrag(X, af);
    mm64_pre(D, af, Y);
}

__global__ __launch_bounds__(256)
void shared_generator_operator_kernel(
    const float* __restrict__ h_i, const float* __restrict__ h_j,
    const float* __restrict__ W1, const float* __restrict__ b1,
    const float* __restrict__ W2, const float* __restrict__ b2,
    const float* __restrict__ W3, const float* __restrict__ b3,
    const float* __restrict__ Wc, const float* __restrict__ G,
    float* __restrict__ out_op,    // [B,64,64]
    float* __restrict__ out_code,  // [B,32]
    float* __restrict__ out_coef,  // [B,16]
    float* __restrict__ out_raw)   // [B,64,64]
{
    __shared__ float Am[F_DIM * LD];   // scaled A (raw before scaling)
    __shared__ float Rm[F_DIM * LD];   // running exp accumulator
    __shared__ float Tm[F_DIM * LD];   // matmul temp
    __shared__ float xbuf[2 * F_DIM];
    __shared__ float h1s[HID];
    __shared__ float h2s[HID];
    __shared__ float codes[CODE];
    __shared__ float coefs[KGEN];
    __shared__ float rowsum[F_DIM];
    __shared__ float scaleS;
    __shared__ int   sS;

    const int tid = threadIdx.x;
    const size_t b = blockIdx.x;

    // ---------------- MLP (threads 0..63) ----------------
    if (tid < F_DIM) {
        xbuf[tid]         = h_i[b * F_DIM + tid];
        xbuf[F_DIM + tid] = h_j[b * F_DIM + tid];
    }
    __syncthreads();
    if (tid < HID) {
        float acc = b1[tid];
        #pragma unroll 8
        for (int j = 0; j < 2 * F_DIM; ++j) acc = fmaf(W1[tid * 2 * F_DIM + j], xbuf[j], acc);
        h1s[tid] = fmaxf(acc, 0.0f);
    }
    __syncthreads();
    if (tid < HID) {
        float acc = b2[tid];
        #pragma unroll 8
        for (int j = 0; j < HID; ++j) acc = fmaf(W2[tid * HID + j], h1s[j], acc);
        h2s[tid] = fmaxf(acc, 0.0f);
    }
    __syncthreads();
    if (tid < CODE) {
        float acc = b3[tid];
        #pragma unroll 8
        for (int j = 0; j < HID; ++j) acc = fmaf(W3[tid * HID + j], h2s[j], acc);
        codes[tid] = acc;
        out_code[b * CODE + tid] = acc;
    }
    __syncthreads();
    if (tid < KGEN) {
        float acc = 0.0f;
        #pragma unroll 8
        for (int j = 0; j < CODE; ++j) acc = fmaf(Wc[tid * CODE + j], codes[j], acc);
        coefs[tid] = acc;
        out_coef[b * KGEN + tid] = acc;
    }
    __syncthreads();

    // ---------------- raw_matrix = sum_k c_k * skew(G_k) ----------------
    for (int e = tid; e < F_DIM * F_DIM; e += 256) {
        const int i = e >> 6, j = e & 63;
        float a = 0.0f;
        #pragma unroll
        for (int k = 0; k < KGEN; ++k) {
            const float g  = G[(size_t)k * F_DIM * F_DIM + i * F_DIM + j];
            const float gt = G[(size_t)k * F_DIM * F_DIM + j * F_DIM + i];
            a = fmaf(coefs[k], 0.5f * (g - gt), a);
        }
        Am[i * LD + j] = a;
        out_raw[b * (F_DIM * F_DIM) + e] = a;
    }
    __syncthreads();

    // ---------------- inf-norm -> scaling exponent s ----------------
    if (tid < F_DIM) {
        float sum = 0.0f;
        for (int j = 0; j < F_DIM; ++j) sum += fabsf(Am[tid * LD + j]);
        rowsum[tid] = sum;
    }
    __syncthreads();
    if (tid == 0) {
        float m = 0.0f;
        for (int i = 0; i < F_DIM; ++i) m = fmaxf(m, rowsum[i]);
        int s = 0; float sc = 1.0f;
        while (m > 1.0f && s < 30) { m *= 0.5f; sc *= 0.5f; ++s; }
        sS = s; scaleS = sc;
    }
    __syncthreads();
    const float sc = scaleS;
    const int   s  = sS;
    for (int e = tid; e < F_DIM * F_DIM; e += 256) {
        Am[(e >> 6) * LD + (e & 63)] *= sc;
    }
    __syncthreads();

    // ---------------- Horner: R = I + A/12; for t=11..1: R = I + (A@R)/t ----
    {
        const float inv = 1.0f / (float)NTAY;
        for (int e = tid; e < F_DIM * F_DIM; e += 256) {
            const int i = e >> 6, j = e & 63;
            Rm[i * LD + j] = ((i == j) ? 1.0f : 0.0f) + Am[i * LD + j] * inv;
        }
    }
    __syncthreads();

    // A-fragments are loop-invariant across the Horner iterations: load once.
    v2f afrag[16];
    load_afrag(Am, afrag);

    // compile-time reciprocal table (avoids v_div chain in the hot loop)
    const float invt[NTAY] = { 0.0f,
        1.0f / 1.0f, 1.0f / 2.0f, 1.0f / 3.0f, 1.0f / 4.0f,  1.0f / 5.0f,
        1.0f / 6.0f, 1.0f / 7.0f, 1.0f / 8.0f, 1.0f / 9.0f, 1.0f / 10.0f,
        1.0f / 11.0f };

    for (int t = NTAY - 1; t >= 1; --t) {
        mm64_pre(Tm, afrag, Rm);     // T = A @ R  (v_wmma_f32_16x16x4_f32)
        __syncthreads();
        const float inv = invt[t];
        for (int e = tid; e < F_DIM * F_DIM; e += 256) {
            const int i = e >> 6, j = e & 63;
            Rm[i * LD + j] = ((i == j) ? 1.0f : 0.0f) + Tm[i * LD + j] * inv;
        }
        __syncthreads();
    }

    // ---------------- squaring: R <- R@R, s times ----------------
    for (int q = 0; q < s; ++q) {
        mm64(Tm, Rm, Rm);
        __syncthreads();
        for (int e = tid; e < F_DIM * F_DIM; e += 256) {
            const int i = e >> 6, j = e & 63;
            Rm[i * LD + j] = Tm[i * LD + j];
        }
        __syncthreads();
    }

    // ---------------- write operator ----------------
    for (int e = tid; e < F_DIM * F_DIM; e += 256) {
        out_op[b * (F_DIM * F_DIM) + e] = Rm[(e >> 6) * LD + (e & 63)];
    }
}

extern "C" void kernel_launch(void* const* d_in, const int* in_sizes, int n_in,
                              void* d_out, int out_size, void* d_ws, size_t ws_size,
                              hipStream_t stream) {
    const float* h_i = (const float*)d_in[0];
    const float* h_j = (const float*)d_in[1];
    const float* W1  = (const float*)d_in[2];
    const float* b1  = (const float*)d_in[3];
    const float* W2  = (const float*)d_in[4];
    const float* b2  = (const float*)d_in[5];
    const float* W3  = (const float*)d_in[6];
    const float* b3  = (const float*)d_in[7];
    const float* Wc  = (const float*)d_in[8];
    const float* G   = (const float*)d_in[9];

    const size_t B = (size_t)in_sizes[0] / F_DIM;

    float* out = (float*)d_out;
    // return order: operator [B,64,64], code [B,32], coefficients [B,16], raw [B,64,64]
    float* out_op   = out;
    float* out_code = out_op + B * F_DIM * F_DIM;
    float* out_coef = out_code + B * CODE;
    float* out_raw  = out_coef + B * KGEN;

    shared_generator_operator_kernel<<<(unsigned)B, 256, 0, stream>>>(
        h_i, h_j, W1, b1, W2, b2, W3, b3, Wc, G,
        out_op, out_code, out_coef, out_raw);
}